// LSRNNBlock_74457553043736
// MI455X (gfx1250) — compile-verified
//
#include <hip/hip_runtime.h>
#include <math.h>

// ---------------------------------------------------------------------------
// LSRNN block for MI455X (gfx1250, wave32):
//   h     = relu(x @ W1 + b1)              GEMM1  M=16384 K=1024 N=4096
//   tr_ri = h @ W2 + b2                    GEMM2  M=16384 K=4096 N=2048
//   out   = exp(i*phi) * cumprod_L((re,im)/|.|)   8192 chains of length 2048
//
// GEMMs: V_WMMA_F32_16X16X4_F32 (full f32 -- phase errors compound over the
// 2048-step cumprod, bf16 matrix ops would drift ~0.1-0.4 rad).
// Double-buffered LDS pipeline; B stored K-pair-interleaved in LDS so each
// WMMA B fragment is a single conflict-free ds_load_b64 (no VGPR shuffles).
// ---------------------------------------------------------------------------

typedef __attribute__((ext_vector_type(2))) float v2f;
typedef __attribute__((ext_vector_type(8))) float v8f;

#define BM 128
#define BN 128
#define BK 32
#define APAD 36            // A row stride (dwords): 144B rows, 16B aligned, conflict-free
#define BSTR (BN * 2 + 32) // 288 dwords: kpair stride; 288 % 64 == 32 -> lane halves on disjoint banks

// C = act(A[M,K] @ B[K,N] + bias), act = relu if relu!=0
// 256 threads (8 waves); block tile 128x128, K-step 32.
// wave (wid&3, wid>>2) computes a 32x64 subtile = 2x4 WMMA 16x16 tiles.
__launch_bounds__(256)
__global__ void gemm_f32_wmma(const float* __restrict__ A,
                              const float* __restrict__ Bm,
                              const float* __restrict__ bias,
                              float* __restrict__ C,
                              int M, int N, int K, int relu)
{
    __shared__ float As[2][BM * APAD];
    __shared__ float Bs[2][(BK / 2) * BSTR];

    const int tid  = threadIdx.x;
    const int wid  = tid >> 5;
    const int lane = tid & 31;
    const int lm   = lane & 15;   // position within 16-lane half
    const int kh   = lane >> 4;   // half-wave id (K split of A/B fragments)

    const int wm = (wid & 3) * 32;   // wave M offset in block tile
    const int wn = (wid >> 2) * 64;  // wave N offset in block tile

    const int m0 = blockIdx.y * BM;
    const int n0 = blockIdx.x * BN;

    v8f zero = {};
    v8f acc[2][4];
#pragma unroll
    for (int i = 0; i < 2; ++i)
#pragma unroll
        for (int j = 0; j < 4; ++j) acc[i][j] = zero;

    // global->LDS mappings (float4 loads, fully coalesced)
    const int a_c4 = tid & 7;    // A: 8 float4 per 32-float row; rows a_r,+32,+64,+96
    const int a_r  = tid >> 3;
    const int b_c4 = tid & 31;   // B: 32 float4 per row; kpair tasks b_u and b_u+8
    const int b_u  = tid >> 5;

    float4 aReg[4];
    float4 bReg[2][2];

    auto gload = [&](int k0) {
#pragma unroll
        for (int i = 0; i < 4; ++i)
            aReg[i] = *(const float4*)(A + (size_t)(m0 + a_r + i * 32) * K + k0 + a_c4 * 4);
#pragma unroll
        for (int t = 0; t < 2; ++t) {
            const int u = b_u + t * 8;
            bReg[t][0] = *(const float4*)(Bm + (size_t)(k0 + 2 * u    ) * N + n0 + b_c4 * 4);
            bReg[t][1] = *(const float4*)(Bm + (size_t)(k0 + 2 * u + 1) * N + n0 + b_c4 * 4);
        }
    };
    auto sstore = [&](int buf) {
#pragma unroll
        for (int i = 0; i < 4; ++i)
            *(float4*)(&As[buf][(a_r + i * 32) * APAD + a_c4 * 4]) = aReg[i];
#pragma unroll
        for (int t = 0; t < 2; ++t) {
            const int u = b_u + t * 8;
            const float4 lo = bReg[t][0], hi = bReg[t][1];
            // K-pair interleave: Bs[u][n*2+0]=B[2u][n], [n*2+1]=B[2u+1][n]
            const float4 p0 = make_float4(lo.x, hi.x, lo.y, hi.y);
            const float4 p1 = make_float4(lo.z, hi.z, lo.w, hi.w);
            *(float4*)(&Bs[buf][u * BSTR + b_c4 * 8    ]) = p0;
            *(float4*)(&Bs[buf][u * BSTR + b_c4 * 8 + 4]) = p1;
        }
    };

    gload(0);
    sstore(0);

    const int nK = K / BK;
    for (int kb = 0; kb < nK; ++kb) {
        __syncthreads();
        const int cur = kb & 1;
        if (kb + 1 < nK) {
            gload((kb + 1) * BK);                       // overlap with WMMA burst below
            if (kb + 2 < nK) {                          // warm L2 two blocks ahead
                __builtin_prefetch(A + (size_t)(m0 + a_r) * K + (kb + 2) * BK + a_c4 * 4, 0, 1);
                __builtin_prefetch(Bm + (size_t)((kb + 2) * BK + 2 * b_u) * N + n0 + b_c4 * 4, 0, 1);
            }
        }

#pragma unroll
        for (int kk = 0; kk < BK; kk += 4) {
            // A frag (16x4 f32, v2f): lane half kh holds K = kk+2kh, kk+2kh+1
            v2f a[2];
#pragma unroll
            for (int i = 0; i < 2; ++i)
                a[i] = *(const v2f*)(&As[cur][(wm + i * 16 + lm) * APAD + kk + 2 * kh]);

            // B frag (4x16 f32, v2f): one aligned b64 per tile thanks to interleave
            v2f b[4];
#pragma unroll
            for (int j = 0; j < 4; ++j)
                b[j] = *(const v2f*)(&Bs[cur][((kk >> 1) + kh) * BSTR + (wn + j * 16 + lm) * 2]);

#pragma unroll
            for (int i = 0; i < 2; ++i)
#pragma unroll
                for (int j = 0; j < 4; ++j)
                    acc[i][j] = __builtin_amdgcn_wmma_f32_16x16x4_f32(
                        false, a[i], false, b[j], (short)0, acc[i][j], false, false);
        }

        if (kb + 1 < nK) sstore((kb + 1) & 1);
    }

    // epilogue: C/D layout: VGPR v, lane -> M = 8*(lane>=16) + v, N = lane&15
#pragma unroll
    for (int j = 0; j < 4; ++j) {
        const int n  = n0 + wn + j * 16 + lm;
        const float bv = bias[n];
#pragma unroll
        for (int i = 0; i < 2; ++i) {
            const int mbase = m0 + wm + i * 16 + kh * 8;
#pragma unroll
            for (int v = 0; v < 8; ++v) {
                float r = acc[i][j][v] + bv;
                if (relu) r = fmaxf(r, 0.0f);
                C[(size_t)(mbase + v) * N + n] = r;
            }
        }
    }
}

// out[b,l,h] = exp(i*phi_h) * prod_{t<=l} tr[b,t,h];  tr = (re,im)/|.| read
// straight from tr_ri[row, h] / tr_ri[row, h+1024] (normalization fused here,
// saving a 256MB round trip). One thread per (b,h) chain; loads/stores are
// coalesced across h at every step.
__global__ void cumprod_kernel(const float* __restrict__ tr_ri,
                               const float* __restrict__ phases,
                               float* __restrict__ out)
{
    const int tid = blockIdx.x * blockDim.x + threadIdx.x; // 0..8191
    const int b = tid >> 10;
    const int h = tid & 1023;
    float s, c;
    __sincosf(phases[h], &s, &c);
    float ar = c, ai = s;
    float2* out2 = (float2*)out;
    for (int l = 0; l < 2048; ++l) {
        const size_t row = (size_t)(b * 2048 + l);
        const float re = tr_ri[row * 2048 + h];
        const float im = tr_ri[row * 2048 + h + 1024];
        const float inv0 = rsqrtf(re * re + im * im);
        const float tr = re * inv0, ti = im * inv0;
        const float nr = ar * tr - ai * ti;
        const float ni = ar * ti + ai * tr;
        const float inv1 = rsqrtf(nr * nr + ni * ni); // keep product on unit circle
        ar = nr * inv1;
        ai = ni * inv1;
        out2[row * 1024 + h] = make_float2(ar, ai);   // complex64 interleaved
    }
}

extern "C" void kernel_launch(void* const* d_in, const int* in_sizes, int n_in,
                              void* d_out, int out_size, void* d_ws, size_t ws_size,
                              hipStream_t stream)
{
    const float* x      = (const float*)d_in[0];
    const float* W1     = (const float*)d_in[1];
    const float* b1     = (const float*)d_in[2];
    const float* W2     = (const float*)d_in[3];
    const float* b2     = (const float*)d_in[4];
    const float* phases = (const float*)d_in[5];
    float* out = (float*)d_out;

    const int M = 16384;   // B*L
    const int D = 1024;
    const int F = 4096;    // 4H
    const int N2 = 2048;   // 2H

    // workspace: [0,256MB) h[M,F] ; [256MB,384MB) tr_ri[M,N2]
    float* h_buf = (float*)d_ws;
    float* tr_ri = h_buf + (size_t)M * F;

    dim3 blk(256);
    gemm_f32_wmma<<<dim3(F / BN, M / BM), blk, 0, stream>>>(x,     W1, b1, h_buf, M, F,  D, 1);
    gemm_f32_wmma<<<dim3(N2 / BN, M / BM), blk, 0, stream>>>(h_buf, W2, b2, tr_ri, M, N2, F, 0);
    cumprod_kernel<<<8192 / 256, 256, 0, stream>>>(tr_ri, phases, out);
}